// DifferentiableRasterizer_34677565948134
// MI455X (gfx1250) — compile-verified
//
#include <hip/hip_runtime.h>

typedef float v2f __attribute__((ext_vector_type(2)));
typedef float v8f __attribute__((ext_vector_type(8)));

#define NSAMP   50
#define NSTROKE 64
#define CANVAS  512

__global__ __launch_bounds__(256)
void DifferentiableRasterizer_wmma_kernel(const float* __restrict__ strokes,
                                          const float* __restrict__ widths,
                                          const float* __restrict__ colors,
                                          float* __restrict__ out)
{
    const int tid  = blockIdx.x * blockDim.x + threadIdx.x;
    const int lane = threadIdx.x & 31;
    const int sub  = lane & 15;
    const bool hi  = lane >= 16;          // lanes 16-31 carry K=2,3 operands
    const int base = tid & ~31;           // first pixel of this wave's 32-pixel strip

    // ---- Pixel operand (B matrix) for the two 16-pixel tiles; built once. ----
    // B column(pixel) = [-2x, -2y, 1, x*x+y*y]^T
    // Layout (4x16 f32): lanes 0-15 -> VGPR0=K0, VGPR1=K1 ; lanes 16-31 -> VGPR0=K2, VGPR1=K3
    v2f Blo, Bhi;
    {
        int p0 = base + sub;
        int p1 = base + 16 + sub;
        float x0 = (float)(p0 & (CANVAS - 1)), y0 = (float)(p0 >> 9);
        float x1 = (float)(p1 & (CANVAS - 1)), y1 = (float)(p1 >> 9);
        Blo.x = hi ? 1.0f                  : -2.0f * x0;
        Blo.y = hi ? (x0 * x0 + y0 * y0)   : -2.0f * y0;
        Bhi.x = hi ? 1.0f                  : -2.0f * x1;
        Bhi.y = hi ? (x1 * x1 + y1 * y1)   : -2.0f * y1;
    }

    const int   p    = base + lane;               // this lane's pixel (after reduction)
    const float INVN = 1.0f / (float)(NSAMP - 1);

    float cr = 1.0f, cg = 1.0f, cb = 1.0f;        // canvas initialized to ones

    for (int st = 0; st < NSTROKE; ++st) {
        const float* cp = strokes + st * 8;       // wave-uniform loads
        const float p0x = cp[0], p0y = cp[1];
        const float p1x = cp[2], p1y = cp[3];
        const float p2x = cp[4], p2y = cp[5];
        const float p3x = cp[6], p3y = cp[7];
        const float w     = widths[st];
        const float col_r = colors[st * 3 + 0];
        const float col_g = colors[st * 3 + 1];
        const float col_b = colors[st * 3 + 2];

        float mlo = 3.0e38f, mhi = 3.0e38f;

        // 4 M-tiles of 16 samples each (samples 50..63 clamp to t=1: min unaffected)
        #pragma unroll
        for (int T = 0; T < 4; ++T) {
            const int   s = T * 16 + sub;         // sample row M for this lane
            const float t = fminf((float)s, (float)(NSAMP - 1)) * INVN;
            const float u = 1.0f - t;
            const float w0 = u * u * u;
            const float w1 = 3.0f * u * u * t;
            const float w2 = 3.0f * u * t * t;
            const float w3 = t * t * t;
            const float px = (w0 * p0x + w1 * p1x + w2 * p2x + w3 * p3x) * (float)CANVAS;
            const float py = (w0 * p0y + w1 * p1y + w2 * p2y + w3 * p3y) * (float)CANVAS;

            // A row(sample) = [px, py, px*px+py*py, 1]
            // Layout (16x4 f32): lanes 0-15 -> VGPR0=K0, VGPR1=K1 ; lanes 16-31 -> K2,K3
            v2f A;
            A.x = hi ? (px * px + py * py) : px;
            A.y = hi ? 1.0f                : py;

            v8f c0 = {};
            // D[sample][pixel] = dist^2 ; one WMMA = 16 samples x 16 pixels
            v8f dlo = __builtin_amdgcn_wmma_f32_16x16x4_f32(
                false, A, false, Blo, (short)0, c0, false, false);
            v8f dhi = __builtin_amdgcn_wmma_f32_16x16x4_f32(
                false, A, false, Bhi, (short)0, c0, false, false);

            #pragma unroll
            for (int r = 0; r < 8; ++r) {
                mlo = fminf(mlo, dlo[r]);
                mhi = fminf(mhi, dhi[r]);
            }
        }

        // lane n holds samples M=0..7, lane n+16 holds M=8..15 for the same pixel col:
        // finish the 16-sample min with one xor-16 shuffle per tile.
        mlo = fminf(mlo, __shfl_xor(mlo, 16, 32));
        mhi = fminf(mhi, __shfl_xor(mhi, 16, 32));

        // lane L<16 -> pixel base+L (tile lo) ; lane L>=16 -> pixel base+L (tile hi)
        const float d2   = hi ? mhi : mlo;
        const float dist = sqrtf(fmaxf(d2, 0.0f));
        const float z    = (w - dist) * 2.0f;
        const float alpha = 1.0f / (1.0f + __expf(-z));   // sigmoid
        const float ia    = 1.0f - alpha;

        cr = cr * ia + col_r * alpha;
        cg = cg * ia + col_g * alpha;
        cb = cb * ia + col_b * alpha;
    }

    const int HW = CANVAS * CANVAS;
    out[0 * HW + p] = cr;
    out[1 * HW + p] = cg;
    out[2 * HW + p] = cb;
}

extern "C" void kernel_launch(void* const* d_in, const int* in_sizes, int n_in,
                              void* d_out, int out_size, void* d_ws, size_t ws_size,
                              hipStream_t stream) {
    (void)in_sizes; (void)n_in; (void)out_size; (void)d_ws; (void)ws_size;
    const float* strokes = (const float*)d_in[0];   // (64,4,2) f32
    const float* widths  = (const float*)d_in[1];   // (64,)    f32
    const float* colors  = (const float*)d_in[2];   // (64,3)   f32
    float* out = (float*)d_out;                     // (1,3,512,512) f32

    const int HW = CANVAS * CANVAS;                 // 262144 pixels, 1 thread each
    dim3 block(256);
    dim3 grid(HW / 256);
    DifferentiableRasterizer_wmma_kernel<<<grid, block, 0, stream>>>(
        strokes, widths, colors, out);
}